// LossModule_69423851372587
// MI455X (gfx1250) — compile-verified
//
#include <hip/hip_runtime.h>
#include <math.h>

typedef __attribute__((ext_vector_type(2))) float v2f;
typedef __attribute__((ext_vector_type(8))) float v8f;

#define D_DIM 256
#define K_DIM 128
#define N_NEG 64
#define T_SEL 16

// Workspace layout (floats): [0] = sum|F F^T - I|, [1..64] = ||neg_n||^2,
//                            [65..192] = ||F_k||^2
#define WS_ORTHO 0
#define WS_NN2   1
#define WS_FN2   65

// ---------------------------------------------------------------------------
// Kernel A: gram = F @ F^T via V_WMMA_F32_16X16X4_F32 -> sum|gram - I| into
// ws[0]; also precompute negative norms and F-row norms.
// Grid: 64 blocks (8x8 tiles of 128x128 gram), 1 wave each.
// ---------------------------------------------------------------------------
__global__ void __launch_bounds__(32)
gram_norms_kernel(const float* __restrict__ F,
                  const float* __restrict__ negatives,
                  float* __restrict__ ws) {
    const int lane = threadIdx.x;        // 0..31
    const int tile = blockIdx.x;         // 0..63
    const int i0   = (tile >> 3) << 4;
    const int j0   = (tile & 7) << 4;
    const int m    = lane & 15;
    const int off  = (lane >> 4) << 1;   // lanes 16-31 hold K=k+2,k+3

    const float* Fa = F + (i0 + m) * D_DIM + off;
    const float* Fb = F + (j0 + m) * D_DIM + off;

    v8f c = {};
    #pragma unroll 4
    for (int k = 0; k < D_DIM; k += 4) {
        v2f a = *(const v2f*)(Fa + k);
        v2f b = *(const v2f*)(Fb + k);
        c = __builtin_amdgcn_wmma_f32_16x16x4_f32(false, a, false, b,
                                                  (short)0, c, false, false);
    }

    const int baseM = (lane >> 4) << 3;
    float local = 0.0f;
    #pragma unroll
    for (int i = 0; i < 8; ++i) {
        const int gi = i0 + baseM + i;
        const int gj = j0 + m;
        local += fabsf(c[i] - ((gi == gj) ? 1.0f : 0.0f));
    }
    #pragma unroll
    for (int s = 16; s; s >>= 1) local += __shfl_xor(local, s, 32);
    if (lane == 0) atomicAdd(&ws[WS_ORTHO], local);

    // negative norms: block n handles negatives[n]
    const int n = blockIdx.x;
    float acc = 0.0f;
    #pragma unroll
    for (int d = lane; d < D_DIM; d += 32) {
        const float x = negatives[n * D_DIM + d];
        acc += x * x;
    }
    #pragma unroll
    for (int s = 16; s; s >>= 1) acc += __shfl_xor(acc, s, 32);
    if (lane == 0) ws[WS_NN2 + n] = acc;

    // F row norms: block n handles rows n and n+64
    #pragma unroll
    for (int t = 0; t < 2; ++t) {
        const int kk = n + t * 64;
        float fa = 0.0f;
        #pragma unroll
        for (int d = lane; d < D_DIM; d += 32) {
            const float x = F[kk * D_DIM + d];
            fa += x * x;
        }
        #pragma unroll
        for (int s = 16; s; s >>= 1) fa += __shfl_xor(fa, s, 32);
        if (lane == 0) ws[WS_FN2 + kk] = fa;
    }
}

// ---------------------------------------------------------------------------
// Kernel B: 16 rows per wave.
//  stage 0: async-stage the 16x128 g tile into LDS (overlapped)
//  stage 1: true_d, ||vhat||^2
//  stage 2: WMMA: vhat_tile @ [negatives^T | F^T]  (12 accumulators, shared A)
//           -> neg_d hinge (Ju) ; cross_F parked in LDS
//  stage 3: parallel top-16-smallest on g (2 lanes/row), d_ft from LDS -> Jt
//  stage 4: out = Ju + Jt + 1e-3 * ortho^2
// ---------------------------------------------------------------------------
__global__ void __launch_bounds__(32)
row_loss_kernel(const float* __restrict__ v,
                const float* __restrict__ vhat,
                const float* __restrict__ g,
                const float* __restrict__ F,
                const float* __restrict__ negatives,
                const float* __restrict__ ws,
                float* __restrict__ out) {
    const int lane = threadIdx.x;
    const int r0   = blockIdx.x << 4;
    const int m    = lane & 15;
    const int half = lane >> 4;

    __shared__ float s_g[16][K_DIM];    // staged g tile
    __shared__ float s_xf[16][K_DIM];   // vhat_tile @ F^T
    __shared__ float s_true[16];
    __shared__ float s_vn2[16];
    __shared__ float s_ju[16];
    __shared__ float s_nn2[N_NEG];
    __shared__ float s_fn2[K_DIM];

    // ---- stage 0: kick off async staging of the g tile (16*128 floats) ----
    {
        const unsigned lds_addr =
            (unsigned)(uintptr_t)(&s_g[0][0]) + (unsigned)(lane * 256);
        const unsigned long long gaddr =
            (unsigned long long)(uintptr_t)(g + (size_t)r0 * K_DIM) +
            (unsigned long long)(lane * 256);
        #pragma unroll
        for (int i = 0; i < 16; ++i) {
            asm volatile("global_load_async_to_lds_b128 %0, %1, off offset:%2"
                         :
                         : "v"(lds_addr), "v"(gaddr), "i"(i * 16)
                         : "memory");
        }
    }

    // ---- stage 1: true_d and vhat row norms (lane pair splits D) ----
    {
        const int row   = r0 + m;
        const float* pv = v    + row * D_DIM + half * (D_DIM / 2);
        const float* ph = vhat + row * D_DIM + half * (D_DIM / 2);
        float st = 0.0f, sv = 0.0f;
        #pragma unroll 8
        for (int d = 0; d < D_DIM / 2; ++d) {
            const float hv = ph[d];
            const float dv = hv - pv[d];
            st += dv * dv;
            sv += hv * hv;
        }
        st += __shfl_xor(st, 16, 32);
        sv += __shfl_xor(sv, 16, 32);
        if (half == 0) {
            s_true[m] = sqrtf(st);
            s_vn2[m]  = sv;
        }
    }
    s_nn2[lane]      = ws[WS_NN2 + lane];
    s_nn2[lane + 32] = ws[WS_NN2 + lane + 32];
    #pragma unroll
    for (int t = 0; t < 4; ++t) s_fn2[lane + t * 32] = ws[WS_FN2 + lane + t * 32];
    __syncthreads();

    // ---- stage 2: WMMA cross products against negatives (4 tiles) + F (8) ----
    const int off = half << 1;
    const float* pa  = vhat      + (r0 + m) * D_DIM + off;
    const float* pb0 = negatives + ( 0 + m) * D_DIM + off;
    const float* pb1 = negatives + (16 + m) * D_DIM + off;
    const float* pb2 = negatives + (32 + m) * D_DIM + off;
    const float* pb3 = negatives + (48 + m) * D_DIM + off;
    const float* pf  = F + m * D_DIM + off;   // + kt*16*D_DIM per tile

    v8f cn0 = {}, cn1 = {}, cn2 = {}, cn3 = {};
    v8f cf0 = {}, cf1 = {}, cf2 = {}, cf3 = {};
    v8f cf4 = {}, cf5 = {}, cf6 = {}, cf7 = {};

#define WMMA_ACC(CACC, PTR) \
    CACC = __builtin_amdgcn_wmma_f32_16x16x4_f32( \
        false, a, false, *(const v2f*)((PTR) + k), (short)0, CACC, false, false)

    for (int k = 0; k < D_DIM; k += 4) {
        v2f a = *(const v2f*)(pa + k);
        WMMA_ACC(cn0, pb0);
        WMMA_ACC(cn1, pb1);
        WMMA_ACC(cn2, pb2);
        WMMA_ACC(cn3, pb3);
        WMMA_ACC(cf0, pf + 0 * 16 * D_DIM);
        WMMA_ACC(cf1, pf + 1 * 16 * D_DIM);
        WMMA_ACC(cf2, pf + 2 * 16 * D_DIM);
        WMMA_ACC(cf3, pf + 3 * 16 * D_DIM);
        WMMA_ACC(cf4, pf + 4 * 16 * D_DIM);
        WMMA_ACC(cf5, pf + 5 * 16 * D_DIM);
        WMMA_ACC(cf6, pf + 6 * 16 * D_DIM);
        WMMA_ACC(cf7, pf + 7 * 16 * D_DIM);
    }
#undef WMMA_ACC

    // C layout: lanes 0-15 -> M=i, N=lane; lanes 16-31 -> M=8+i, N=lane-16
    const int baseM = half << 3;

    // Ju hinge accumulation from the negative tiles
    float hj[8];
    #pragma unroll
    for (int i = 0; i < 8; ++i) hj[i] = 0.0f;
    auto accum = [&](const v8f& c, int nt) {
        const float nn2 = s_nn2[nt * 16 + m];
        #pragma unroll
        for (int i = 0; i < 8; ++i) {
            const int M    = baseM + i;
            const float d2 = s_vn2[M] + nn2 - 2.0f * c[i];
            const float nd = sqrtf(fmaxf(d2, 0.0f));
            hj[i] += fmaxf(0.0f, 1.0f + s_true[M] - nd);
        }
    };
    accum(cn0, 0); accum(cn1, 1); accum(cn2, 2); accum(cn3, 3);
    #pragma unroll
    for (int i = 0; i < 8; ++i) {
        float h = hj[i];
        h += __shfl_xor(h, 1, 32);
        h += __shfl_xor(h, 2, 32);
        h += __shfl_xor(h, 4, 32);
        h += __shfl_xor(h, 8, 32);
        if (m == 0) s_ju[baseM + i] = h;
    }

    // park vhat @ F^T in LDS
    auto store_xf = [&](const v8f& c, int kt) {
        #pragma unroll
        for (int i = 0; i < 8; ++i) s_xf[baseM + i][kt * 16 + m] = c[i];
    };
    store_xf(cf0, 0); store_xf(cf1, 1); store_xf(cf2, 2); store_xf(cf3, 3);
    store_xf(cf4, 4); store_xf(cf5, 5); store_xf(cf6, 6); store_xf(cf7, 7);

    // g tile must be resident before selection
    asm volatile("s_wait_asynccnt 0x0" ::: "memory");
    __syncthreads();

    // ---- stage 3: top-16-smallest of g row; 2 lanes per row ----
    // lane m scans j in [0,64), lane m+16 scans j in [64,128)
    float gvals[T_SEL];
    int   idxs[T_SEL];
    float gsum = 0.0f;
    {
        const float* grow = &s_g[m][half * 64];
        const int jbase   = half * 64;
        unsigned used0 = 0u, used1 = 0u;
        for (int t = 0; t < T_SEL; ++t) {
            float best = 3.402823466e38f;
            int   bj   = jbase;
            for (int j = 0; j < 64; ++j) {
                const bool u = (j < 32) ? ((used0 >> j) & 1u)
                                        : ((used1 >> (j - 32)) & 1u);
                if (u) continue;
                const float val = grow[j];
                if (val < best) { best = val; bj = jbase + j; }
            }
            const float obest = __shfl_xor(best, 16, 32);
            const int   obj   = __shfl_xor(bj, 16, 32);
            const bool take_other =
                (obest < best) || ((obest == best) && (obj < bj));
            if (!take_other) {
                const int lj = bj - jbase;
                if (lj < 32) used0 |= 1u << lj;
                else         used1 |= 1u << (lj - 32);
            }
            const float wv = take_other ? obest : best;
            const int   wj = take_other ? obj   : bj;
            gvals[t] = wv;
            idxs[t]  = wj;
            gsum    += wv;
        }
    }

    // ---- stage 4: Jt from LDS cross-products; final sum ----
    if (lane < 16) {
        const float inv = 1.0f / (gsum + 1e-10f);
        const float td  = s_true[lane];
        const float vn2 = s_vn2[lane];
        float jt = 0.0f;
        #pragma unroll
        for (int t = 0; t < T_SEL; ++t) {
            const float gt  = gvals[t] * inv;
            const float om  = 1.0f - gt;
            const float mt  = om * om;                    // MARGIN = 1.0
            const int   kk  = idxs[t];
            const float d2  = vn2 + s_fn2[kk] - 2.0f * s_xf[lane][kk];
            const float dft = sqrtf(fmaxf(d2, 0.0f));
            jt += fmaxf(0.0f, mt + td - dft);
        }
        jt *= (1.0f / (float)T_SEL);

        const float ju    = s_ju[lane] * (1.0f / (float)N_NEG);
        const float ortho = ws[WS_ORTHO];
        out[r0 + lane] = ju + jt + 1e-3f * ortho * ortho;
    }
}

// ---------------------------------------------------------------------------
extern "C" void kernel_launch(void* const* d_in, const int* in_sizes, int n_in,
                              void* d_out, int out_size, void* d_ws, size_t ws_size,
                              hipStream_t stream) {
    const float* v         = (const float*)d_in[0];
    const float* vhat      = (const float*)d_in[1];
    const float* g         = (const float*)d_in[2];
    const float* F         = (const float*)d_in[3];
    const float* negatives = (const float*)d_in[4];
    float* out = (float*)d_out;
    float* ws  = (float*)d_ws;

    const int B = in_sizes[0] / D_DIM;   // 8192

    hipMemsetAsync(ws, 0, sizeof(float), stream);

    gram_norms_kernel<<<(K_DIM / 16) * (K_DIM / 16), 32, 0, stream>>>(F, negatives, ws);
    row_loss_kernel<<<B / 16, 32, 0, stream>>>(v, vhat, g, F, negatives, ws, out);
}